// GeolocalizationLoss_36352603194084
// MI455X (gfx1250) — compile-verified
//
#include <hip/hip_runtime.h>
#include <math.h>

// Problem shape (fixed by the reference setup).
#define NG 32768   // geocells
#define NB 2048    // batch rows
#define BLK 256    // 8 wave32 per block

#define DEG2RAD 0.017453292519943295f
#define INV_T   (1.0f / 75.0f)
#define EARTH_R 6371.0f

#if defined(__gfx1250__) && __has_builtin(__builtin_amdgcn_global_load_async_to_lds_b128)
#define USE_ASYNC_LDS 1
#else
#define USE_ASYNC_LDS 0
#endif

// ---------------- block reductions (deterministic, LDS tree) ----------------
__device__ __forceinline__ float block_reduce_max(float v, float* red) {
    const int tid = threadIdx.x;
    red[tid] = v;
    __syncthreads();
    for (int off = BLK / 2; off > 0; off >>= 1) {
        if (tid < off) red[tid] = fmaxf(red[tid], red[tid + off]);
        __syncthreads();
    }
    float r = red[0];
    __syncthreads();
    return r;
}

__device__ __forceinline__ float block_reduce_sum(float v, float* red) {
    const int tid = threadIdx.x;
    red[tid] = v;
    __syncthreads();
    for (int off = BLK / 2; off > 0; off >>= 1) {
        if (tid < off) red[tid] += red[tid + off];
        __syncthreads();
    }
    float r = red[0];
    __syncthreads();
    return r;
}

// ---------------- precompute per-geocell trig (runs once, tiny) -------------
__global__ void geo_pre_g(const float* __restrict__ centroids,
                          float* __restrict__ g_hphi,
                          float* __restrict__ g_hlam,
                          float* __restrict__ g_cos) {
    int g = blockIdx.x * blockDim.x + threadIdx.x;
    if (g >= NG) return;
    float lam = centroids[2 * g] * DEG2RAD;
    float phi = centroids[2 * g + 1] * DEG2RAD;
    g_hphi[g] = 0.5f * phi;
    g_hlam[g] = 0.5f * lam;
    g_cos[g]  = cosf(phi);
}

// ------------- precompute per-row trig + true distance (tiny) ---------------
__global__ void geo_pre_b(const float* __restrict__ coords,
                          const int* __restrict__ cell,
                          const float* __restrict__ centroids,
                          float* __restrict__ b_hphi,
                          float* __restrict__ b_hlam,
                          float* __restrict__ b_cos,
                          float* __restrict__ b_dtrue) {
    int i = blockIdx.x * blockDim.x + threadIdx.x;
    if (i >= NB) return;
    float lam1 = coords[2 * i] * DEG2RAD;
    float phi1 = coords[2 * i + 1] * DEG2RAD;
    float c1 = cosf(phi1);
    b_hphi[i] = 0.5f * phi1;
    b_hlam[i] = 0.5f * lam1;
    b_cos[i]  = c1;
    int j = cell[i];
    float lam2 = centroids[2 * j] * DEG2RAD;
    float phi2 = centroids[2 * j + 1] * DEG2RAD;
    float sp = __sinf(0.5f * (phi2 - phi1));
    float sl = __sinf(0.5f * (lam2 - lam1));
    float a = fmaf(sp, sp, (c1 * cosf(phi2)) * (sl * sl));
    a = fminf(fmaxf(a, 0.0f), 1.0f);
    b_dtrue[i] = (2.0f * EARTH_R) * atan2f(sqrtf(a), sqrtf(1.0f - a));
}

// per-element loss term:  w * (x - m - logZ),  w = exp((d_true - d)/T)
// d = 2*R*atan2(sqrt(a), sqrt(1-a)) is folded as  w = exp(fma(atan2, A1, A0)),
// A1 = -2R/T, A0 = d_true/T.  (x - mlz) < 0 strictly, so w=inf gives -inf, never NaN,
// matching the fp32 reference's overflow behavior.
__device__ __forceinline__ float geo_term(float x, float hpg, float hlg, float cgg,
                                          float hpb, float hlb, float cb,
                                          float A0, float A1, float mlz) {
    float sp = __sinf(hpg - hpb);
    float sl = __sinf(hlg - hlb);
    float a  = fmaf(sp, sp, (cb * cgg) * (sl * sl));
    a = fminf(fmaxf(a, 0.0f), 1.0f);
    float t  = atan2f(sqrtf(a), sqrtf(fmaxf(1.0f - a, 0.0f)));
    float w  = __expf(fmaf(t, A1, A0));
    return w * (x - mlz);
}

// --------------------------- main: one block per row ------------------------
__global__ __launch_bounds__(BLK) void geo_row_kernel(
    const float* __restrict__ outputs,
    const float* __restrict__ g_hphi, const float* __restrict__ g_hlam,
    const float* __restrict__ g_cos,
    const float* __restrict__ b_hphi, const float* __restrict__ b_hlam,
    const float* __restrict__ b_cos,  const float* __restrict__ b_dtrue,
    float* __restrict__ row_loss) {
    extern __shared__ float xrow[];        // 32768 floats = 128 KB (fits 320 KB LDS)
    __shared__ float red[BLK];

    const int b   = blockIdx.x;
    const int tid = threadIdx.x;
    const float* xg = outputs + (size_t)b * NG;

    // ---- stage the whole 128 KB row into LDS: HBM is touched exactly once ----
#if USE_ASYNC_LDS
    {
        typedef int v4i_t __attribute__((ext_vector_type(4)));
        typedef __attribute__((address_space(1))) v4i_t* gvec_p;   // HIP "__device__" AS1
        typedef __attribute__((address_space(3))) v4i_t* lvec_p;   // LDS AS3
        const char* gsrc = (const char*)xg;
        char*       ldst = (char*)xrow;
        for (int i = tid * 16; i < NG * 4; i += BLK * 16) {
            __builtin_amdgcn_global_load_async_to_lds_b128(
                (gvec_p)(gsrc + i), (lvec_p)(ldst + i), 0, 0);
        }
#if __has_builtin(__builtin_amdgcn_s_wait_asynccnt)
        __builtin_amdgcn_s_wait_asynccnt(0);
#else
        asm volatile("s_wait_asynccnt 0" ::: "memory");
#endif
    }
#else
    {
        const float4* src = (const float4*)xg;
        float4*       dst = (float4*)xrow;
        for (int i = tid; i < NG / 4; i += BLK) dst[i] = src[i];
    }
#endif
    __syncthreads();

    const float4* x4 = (const float4*)xrow;

    // ---- pass 1: row max (LDS only) ----
    float m = -INFINITY;
    for (int i = tid; i < NG / 4; i += BLK) {
        float4 v = x4[i];
        m = fmaxf(m, fmaxf(fmaxf(v.x, v.y), fmaxf(v.z, v.w)));
    }
    m = block_reduce_max(m, red);

    // ---- pass 2: Z = sum exp(x - m) (LDS only) ----
    float z = 0.0f;
    for (int i = tid; i < NG / 4; i += BLK) {
        float4 v = x4[i];
        z += __expf(v.x - m) + __expf(v.y - m) + __expf(v.z - m) + __expf(v.w - m);
    }
    z = block_reduce_sum(z, red);
    const float mlz = m + __logf(z);     // m + log Z  (z in [1, 32768], well conditioned)

    // ---- row constants ----
    const float hpb = b_hphi[b];
    const float hlb = b_hlam[b];
    const float cb  = b_cos[b];
    const float A0  = b_dtrue[b] * INV_T;
    const float A1  = -(2.0f * EARTH_R) * INV_T;

    // ---- pass 3: sum_g w * (x - m - logZ); trig tables stream from L2 ----
    const float4* hp4 = (const float4*)g_hphi;
    const float4* hl4 = (const float4*)g_hlam;
    const float4* cg4 = (const float4*)g_cos;
    float acc = 0.0f;
    for (int i = tid; i < NG / 4; i += BLK) {
        __builtin_prefetch(hp4 + i + BLK, 0, 0);   // global_prefetch_b8 (L2-resident trig)
        __builtin_prefetch(hl4 + i + BLK, 0, 0);
        __builtin_prefetch(cg4 + i + BLK, 0, 0);
        float4 xv = x4[i];
        float4 hp = hp4[i];
        float4 hl = hl4[i];
        float4 cg = cg4[i];
        acc += geo_term(xv.x, hp.x, hl.x, cg.x, hpb, hlb, cb, A0, A1, mlz);
        acc += geo_term(xv.y, hp.y, hl.y, cg.y, hpb, hlb, cb, A0, A1, mlz);
        acc += geo_term(xv.z, hp.z, hl.z, cg.z, hpb, hlb, cb, A0, A1, mlz);
        acc += geo_term(xv.w, hp.w, hl.w, cg.w, hpb, hlb, cb, A0, A1, mlz);
    }
    acc = block_reduce_sum(acc, red);
    if (tid == 0) row_loss[b] = -acc;    // -sum(smoothed * log_probs) for this row
}

// --------------------- deterministic final mean over rows -------------------
__global__ __launch_bounds__(BLK) void geo_finalize(const float* __restrict__ row_loss,
                                                    float* __restrict__ out) {
    __shared__ float red[BLK];
    const int tid = threadIdx.x;
    float s = 0.0f;
    for (int i = tid; i < NB; i += BLK) s += row_loss[i];
    red[tid] = s;
    __syncthreads();
    for (int off = BLK / 2; off > 0; off >>= 1) {
        if (tid < off) red[tid] += red[tid + off];
        __syncthreads();
    }
    if (tid == 0) out[0] = red[0] * (1.0f / (float)NB);
}

extern "C" void kernel_launch(void* const* d_in, const int* in_sizes, int n_in,
                              void* d_out, int out_size, void* d_ws, size_t ws_size,
                              hipStream_t stream) {
    const float* outputs     = (const float*)d_in[0];  // [B,G] fp32
    const float* true_coords = (const float*)d_in[1];  // [B,2] fp32
    const float* centroids   = (const float*)d_in[2];  // [G,2] fp32
    const int*   true_cell   = (const int*)d_in[3];    // [B]   int32

    // workspace layout (floats): 3*NG + 5*NB = 108,544 floats (~424 KB)
    float* ws      = (float*)d_ws;
    float* g_hphi  = ws;
    float* g_hlam  = ws + NG;
    float* g_cos   = ws + 2 * NG;
    float* b_hphi  = ws + 3 * NG;
    float* b_hlam  = ws + 3 * NG + NB;
    float* b_cos   = ws + 3 * NG + 2 * NB;
    float* b_dtrue = ws + 3 * NG + 3 * NB;
    float* rloss   = ws + 3 * NG + 4 * NB;

    geo_pre_g<<<(NG + BLK - 1) / BLK, BLK, 0, stream>>>(centroids, g_hphi, g_hlam, g_cos);
    geo_pre_b<<<(NB + BLK - 1) / BLK, BLK, 0, stream>>>(true_coords, true_cell, centroids,
                                                        b_hphi, b_hlam, b_cos, b_dtrue);
    geo_row_kernel<<<NB, BLK, NG * sizeof(float), stream>>>(
        outputs, g_hphi, g_hlam, g_cos, b_hphi, b_hlam, b_cos, b_dtrue, rloss);
    geo_finalize<<<1, BLK, 0, stream>>>(rloss, (float*)d_out);
}